// GraphAttention_pynq_30159260352619
// MI455X (gfx1250) — compile-verified
//
#include <hip/hip_runtime.h>
#include <math.h>

#define F_IN      128
#define HC        128   // H * C
#define NH        4     // heads
#define NEG_SLOPE 0.2f

typedef __attribute__((ext_vector_type(2))) float        v2f;
typedef __attribute__((ext_vector_type(8))) float        v8f;
typedef __attribute__((ext_vector_type(4))) unsigned int v4u;
typedef __attribute__((ext_vector_type(8))) int          v8i;
typedef __attribute__((ext_vector_type(4))) int          v4i;

// ---------------------------------------------------------------------------
// CDNA5 helpers
// ---------------------------------------------------------------------------

// Hardware f32 atomic add at device scope (single VMEM atomic, no CAS loop).
__device__ __forceinline__ void global_atomic_add_f32(float* p, float v) {
    unsigned long long a = (unsigned long long)(uintptr_t)p;
    asm volatile("global_atomic_add_f32 %0, %1, off scope:SCOPE_DEV"
                 :: "v"(a), "v"(v) : "memory");
}

// Float atomic max via monotone int punning (works for all finite floats and -inf).
__device__ __forceinline__ void atomic_max_f32(float* p, float v) {
    if (v >= 0.0f) atomicMax((int*)p,          __float_as_int(v));
    else           atomicMin((unsigned int*)p, __float_as_uint(v));
}

// ---------------------------------------------------------------------------
// K0: init out = 0, m = -inf, den = 0
// ---------------------------------------------------------------------------
__global__ void k_init(float* __restrict__ out, float* __restrict__ m,
                       float* __restrict__ den, int n) {
    int i = blockIdx.x * blockDim.x + threadIdx.x;
    if (i < n * HC) out[i] = 0.0f;
    if (i < n * NH) { m[i] = -INFINITY; den[i] = 0.0f; }
}

// ---------------------------------------------------------------------------
// K1: h = x @ W via V_WMMA_F32_16X16X4_F32.
// Block = 128 threads (4 wave32 waves). Each block computes a 16x128 slab of h.
// W (128x128 f32 = 64KB) is staged into LDS by BOTH CDNA5 DMA paths:
//   - rows  0..63  : Tensor Data Mover (TENSOR_LOAD_TO_LDS, wave 0, TENSORcnt)
//   - rows 64..127 : GLOBAL_LOAD_ASYNC_TO_LDS_B128 (all threads, ASYNCcnt)
// A-frag layout (16x4 f32): lanes 0-15 hold M=lane, K={k0,k0+1};
//                           lanes 16-31 hold M=lane-16, K={k0+2,k0+3}.
// C/D layout: VGPR r -> row r (+8 for lanes>=16), col = lane&15.
// Each wave accumulates TWO 16x16 column tiles (dual accumulators -> A reuse
// and two independent WMMA dependency chains).
// ---------------------------------------------------------------------------
__global__ void k_gemm_wmma(const float* __restrict__ x, const float* __restrict__ W,
                            float* __restrict__ h, int n) {
    __shared__ float Wlds[F_IN * HC];   // 64 KB
    const int tid  = threadIdx.x;
    const int wave = tid >> 5;
    const int lane = tid & 31;

    // --- TDM: rows 0..63 of W (128 wide x 64 high, f32) --------------------
    if (wave == 0) {
        const unsigned           lds_base = (unsigned)(uintptr_t)Wlds;
        const unsigned long long ga       = (unsigned long long)(uintptr_t)W;
        v4u g0;
        g0.x = 1u;                                            // count=1 (valid user D#)
        g0.y = lds_base;                                      // lds_addr
        g0.z = (unsigned)ga;                                  // global_addr[31:0]
        g0.w = (unsigned)((ga >> 32) & 0x01FFFFFFu)           // global_addr[56:32]
             | (2u << 30);                                    // type = 2 ("image")
        v8i g1;
        g1[0] = 0x2 << 16;        // data_size = 2 -> 4 bytes
        g1[1] = 128 << 16;        // tensor_dim0 = 128 (low 16 bits)
        g1[2] = 128 << 16;        // tensor_dim0 hi=0 | tensor_dim1 = 128 (low 16)
        g1[3] = 128 << 16;        // tensor_dim1 hi=0 | tile_dim0 = 128
        g1[4] = 64;               // tile_dim1 = 64 rows | tile_dim2 = 0
        g1[5] = 128;              // tensor_dim0_stride = 128 (low 32)
        g1[6] = 0;                // stride0 hi | tensor_dim1_stride lo
        g1[7] = 0;
        v4i g2 = {}, g3 = {};     // 2D tensor: groups 2/3 unused
        v8i g4 = {};              // extra group (6-arg builtin form)
        __builtin_amdgcn_tensor_load_to_lds(g0, g1, g2, g3, g4, 0);
    }

    // --- Async copies: rows 64..127 (32 KB, 2048 b128 ops, 16 per thread) --
    for (int i = 0; i < 16; ++i) {
        const int f = 8192 + (i * 128 + tid) * 4;             // float index (16B aligned)
        const unsigned lds_off = (unsigned)(uintptr_t)(Wlds + f);
        const unsigned long long ga = (unsigned long long)(uintptr_t)(W + f);
        asm volatile("global_load_async_to_lds_b128 %0, %1, off"
                     :: "v"(lds_off), "v"(ga) : "memory");
    }
    asm volatile("s_wait_asynccnt 0" ::: "memory");
    if (wave == 0) __builtin_amdgcn_s_wait_tensorcnt(0);
    __syncthreads();

    const int row0  = blockIdx.x * 16;
    const int lrow  = lane & 15;
    const int khalf = (lane >> 4) << 1;               // 0 or 2
    int arow = row0 + lrow; if (arow > n - 1) arow = n - 1;
    const float* xrow = x + (size_t)arow * F_IN + khalf;

    // Each wave owns columns [wave*32, wave*32+32): two 16-wide tiles.
    const int col0 = wave * 32;
    const int col1 = col0 + 16;
    v8f acc0 = {}, acc1 = {};
#pragma unroll 8
    for (int k0 = 0; k0 < F_IN; k0 += 4) {
        v2f a, b0, b1;
        a.x  = xrow[k0];
        a.y  = xrow[k0 + 1];
        b0.x = Wlds[(k0 + khalf)     * HC + col0 + lrow];
        b0.y = Wlds[(k0 + khalf + 1) * HC + col0 + lrow];
        b1.x = Wlds[(k0 + khalf)     * HC + col1 + lrow];
        b1.y = Wlds[(k0 + khalf + 1) * HC + col1 + lrow];
        acc0 = __builtin_amdgcn_wmma_f32_16x16x4_f32(
                   false, a, false, b0, (short)0, acc0, false, false);
        acc1 = __builtin_amdgcn_wmma_f32_16x16x4_f32(
                   false, a, false, b1, (short)0, acc1, false, false);
    }

    const int rbase = row0 + ((lane >> 4) << 3);
    float* hp0 = h + (size_t)rbase * HC + col0 + lrow;
    float* hp1 = h + (size_t)rbase * HC + col1 + lrow;
    if (row0 + 16 <= n) {            // uniform fast path: full tile, no guards
#pragma unroll
        for (int r = 0; r < 8; ++r) {
            hp0[(size_t)r * HC] = acc0[r];
            hp1[(size_t)r * HC] = acc1[r];
        }
    } else {                          // tail block only
#pragma unroll
        for (int r = 0; r < 8; ++r) {
            if (rbase + r < n) {
                hp0[(size_t)r * HC] = acc0[r];
                hp1[(size_t)r * HC] = acc1[r];
            }
        }
    }
}

// ---------------------------------------------------------------------------
// K2: per-node attention logits  a_src[n,h] = <h[n,h,:], att_src[h,:]>, same for dst.
// One thread per (node, head): 8 float4 loads, dual dot product.
// ---------------------------------------------------------------------------
__global__ void k_logits(const float* __restrict__ h, const float* __restrict__ att_src,
                         const float* __restrict__ att_dst, float* __restrict__ a_src,
                         float* __restrict__ a_dst, int n) {
    int i = blockIdx.x * blockDim.x + threadIdx.x;  // node*NH + head
    if (i >= n * NH) return;
    const int node = i >> 2, head = i & 3;
    const float4* hp = (const float4*)(h + (size_t)node * HC + head * 32);
    const float4* as = (const float4*)(att_src + head * 32);
    const float4* ad = (const float4*)(att_dst + head * 32);
    float s = 0.0f, d = 0.0f;
#pragma unroll
    for (int j = 0; j < 8; ++j) {
        const float4 hv = hp[j], av = as[j], bv = ad[j];
        s += hv.x * av.x + hv.y * av.y + hv.z * av.z + hv.w * av.w;
        d += hv.x * bv.x + hv.y * bv.y + hv.z * bv.z + hv.w * bv.w;
    }
    a_src[i] = s;
    a_dst[i] = d;
}

__device__ __forceinline__ float4 edge_logit4(const float* a_src, const float* a_dst,
                                              int s, int d) {
    const float4 as = *(const float4*)(a_src + (size_t)s * NH);
    const float4 ad = *(const float4*)(a_dst + (size_t)d * NH);
    float4 e;
    e.x = as.x + ad.x; e.x = e.x > 0.0f ? e.x : e.x * NEG_SLOPE;
    e.y = as.y + ad.y; e.y = e.y > 0.0f ? e.y : e.y * NEG_SLOPE;
    e.z = as.z + ad.z; e.z = e.z > 0.0f ? e.z : e.z * NEG_SLOPE;
    e.w = as.w + ad.w; e.w = e.w > 0.0f ? e.w : e.w * NEG_SLOPE;
    return e;
}

// ---------------------------------------------------------------------------
// K3: segment max  m[dst,h] = max over incoming edges of leaky(e).
// One thread per edge, all 4 heads (float4 loads, 4 atomic maxes).
// ---------------------------------------------------------------------------
__global__ void k_edge_max(const int* __restrict__ ei, const float* __restrict__ a_src,
                           const float* __restrict__ a_dst, float* __restrict__ m, int nE) {
    int e = blockIdx.x * blockDim.x + threadIdx.x;
    if (e >= nE) return;
    const int s = ei[e], d = ei[nE + e];
    const float4 ev = edge_logit4(a_src, a_dst, s, d);
    float* mp = m + (size_t)d * NH;
    atomic_max_f32(mp + 0, ev.x);
    atomic_max_f32(mp + 1, ev.y);
    atomic_max_f32(mp + 2, ev.z);
    atomic_max_f32(mp + 3, ev.w);
}

// ---------------------------------------------------------------------------
// K4: segment sum  den[dst,h] = sum exp(e - m[dst,h]).
// ---------------------------------------------------------------------------
__global__ void k_edge_den(const int* __restrict__ ei, const float* __restrict__ a_src,
                           const float* __restrict__ a_dst, const float* __restrict__ m,
                           float* __restrict__ den, int nE) {
    int e = blockIdx.x * blockDim.x + threadIdx.x;
    if (e >= nE) return;
    const int s = ei[e], d = ei[nE + e];
    const float4 ev = edge_logit4(a_src, a_dst, s, d);
    const float4 mv = *(const float4*)(m + (size_t)d * NH);
    float* dp = den + (size_t)d * NH;
    global_atomic_add_f32(dp + 0, __expf(ev.x - mv.x));
    global_atomic_add_f32(dp + 1, __expf(ev.y - mv.y));
    global_atomic_add_f32(dp + 2, __expf(ev.z - mv.z));
    global_atomic_add_f32(dp + 3, __expf(ev.w - mv.w));
}

// ---------------------------------------------------------------------------
// K5: weighted scatter-add aggregation.
// One wave32 per edge; lane j carries channels [4j,4j+3] (head = j>>3).
// Gather h[src] row (512B/wave coalesced float4s, L2-resident) and scatter
// attn * h[src] into out[dst] with hardware f32 atomics.
// ---------------------------------------------------------------------------
__global__ void k_aggregate(const int* __restrict__ ei, const float* __restrict__ a_src,
                            const float* __restrict__ a_dst, const float* __restrict__ m,
                            const float* __restrict__ den, const float* __restrict__ h,
                            float* __restrict__ out, int nE) {
    const int gid  = blockIdx.x * blockDim.x + threadIdx.x;
    const int e    = gid >> 5;
    const int lane = gid & 31;
    if (e >= nE) return;
    const int s = ei[e];
    const int d = ei[nE + e];
    const int head = lane >> 3;

    float ev = a_src[(size_t)s * NH + head] + a_dst[(size_t)d * NH + head];
    ev = ev > 0.0f ? ev : ev * NEG_SLOPE;
    const float attn = __expf(ev - m[(size_t)d * NH + head]) / den[(size_t)d * NH + head];

    const float4 hv = *(const float4*)(h + (size_t)s * HC + lane * 4);
    float* op = out + (size_t)d * HC + lane * 4;
    __builtin_prefetch(op, 1, 3);   // global_prefetch_b8 on the scatter target
    global_atomic_add_f32(op + 0, attn * hv.x);
    global_atomic_add_f32(op + 1, attn * hv.y);
    global_atomic_add_f32(op + 2, attn * hv.z);
    global_atomic_add_f32(op + 3, attn * hv.w);
}

// ---------------------------------------------------------------------------
// K6: out = relu(out + bias)   (relu flag read on-device; graph-capture safe)
// ---------------------------------------------------------------------------
__global__ void k_epilogue(float* __restrict__ out, const float* __restrict__ bias,
                           const int* __restrict__ relu_flag, int n) {
    int i = blockIdx.x * blockDim.x + threadIdx.x;
    if (i >= n * HC) return;
    float v = out[i] + bias[i & (HC - 1)];
    if (*relu_flag) v = fmaxf(v, 0.0f);
    out[i] = v;
}

// ---------------------------------------------------------------------------
// Launch
// ---------------------------------------------------------------------------
extern "C" void kernel_launch(void* const* d_in, const int* in_sizes, int n_in,
                              void* d_out, int out_size, void* d_ws, size_t ws_size,
                              hipStream_t stream) {
    const int*   relu_flag = (const int*)d_in[2];
    const float* x         = (const float*)d_in[3];
    const int*   ei        = (const int*)d_in[4];
    const float* W         = (const float*)d_in[6];
    const float* att_src   = (const float*)d_in[7];
    const float* att_dst   = (const float*)d_in[8];
    const float* bias      = (const float*)d_in[9];

    const int n = in_sizes[3] / F_IN;   // nodes
    const int e = in_sizes[4] / 2;      // edges

    // Workspace layout (floats): h[N*128] | a_src[N*4] | a_dst[N*4] | m[N*4] | den[N*4]
    float* h     = (float*)d_ws;
    float* a_src = h     + (size_t)n * HC;
    float* a_dst = a_src + (size_t)n * NH;
    float* mbuf  = a_dst + (size_t)n * NH;
    float* den   = mbuf  + (size_t)n * NH;
    float* out   = (float*)d_out;

    const int B = 256;
    k_init     <<<(n * HC + B - 1) / B, B, 0, stream>>>(out, mbuf, den, n);
    k_gemm_wmma<<<(n + 15) / 16, 128, 0, stream>>>(x, W, h, n);
    k_logits   <<<(n * NH + B - 1) / B, B, 0, stream>>>(h, att_src, att_dst, a_src, a_dst, n);
    k_edge_max <<<(e + B - 1) / B, B, 0, stream>>>(ei, a_src, a_dst, mbuf, e);
    k_edge_den <<<(e + B - 1) / B, B, 0, stream>>>(ei, a_src, a_dst, mbuf, den, e);
    k_aggregate<<<((size_t)e * 32 + B - 1) / B, B, 0, stream>>>(ei, a_src, a_dst, mbuf, den, h, out, e);
    k_epilogue <<<(n * HC + B - 1) / B, B, 0, stream>>>(out, bias, relu_flag, n);
}